// TSPDecoder_89550068122221
// MI455X (gfx1250) — compile-verified
//
#include <hip/hip_runtime.h>
#include <math.h>

// ---------------------------------------------------------------------------
// TXL decoder step (qlen=1) for MI455X / gfx1250, wave32.
//  * Dense GEMMs: V_WMMA_F32_16X16X4_F32, one wave -> 16x32 C strip (A fragment
//    reused across two B tiles).
//  * Cross-attention streams h_enc once per layer (3 GB total -> ~130us HBM
//    floor at 23.3 TB/s), with a 3-deep GLOBAL_LOAD_ASYNC_TO_LDS pipeline
//    (ASYNCcnt-tracked, 4-slot LDS ring) when the toolchain exposes it.
//  * rel_shift(qlen==1) == identity; tanh-clip bounds scores -> single-pass
//    softmax with no running max.
//
// Assumed d_in flattening (compile-only loop; documented):
//   [0]=h_t (1,256,512) f32   [1]=h_enc (1024,256,512) f32   [2]=mask (1,1024,256) bool/u8
//   [3..170]  params.layers[0..5], each 28 tensors in sorted-key order (enum below)
//   [171]=u (16,32)  [172]=v (16,32)   [173..178]=mem0..mem5 (64,256,512) f32
// ---------------------------------------------------------------------------

#define DM     512
#define NH     16
#define DH     32
#define BB     256
#define NN     1024
#define MEMLEN 64
#define NLAYER 6

typedef float v2f __attribute__((ext_vector_type(2)));
typedef float v8f __attribute__((ext_vector_type(8)));

#if defined(__has_builtin)
#if __has_builtin(__builtin_amdgcn_global_load_async_to_lds_b32) && \
    __has_builtin(__builtin_amdgcn_s_wait_asynccnt)
#define USE_ASYNC_LDS 1
#endif
#endif
#ifndef USE_ASYNC_LDS
#define USE_ASYNC_LDS 0
#endif

#if USE_ASYNC_LDS
typedef __attribute__((address_space(1))) int g_int;   // global
typedef __attribute__((address_space(3))) int l_int;   // LDS
__device__ __forceinline__ void async_copy_b32(float* lds_dst, const float* gsrc) {
  float* gs = const_cast<float*>(gsrc);
  __builtin_amdgcn_global_load_async_to_lds_b32(
      (g_int*)gs, (l_int*)lds_dst, 0, 0);
}
#endif

__device__ __forceinline__ float wave_sum(float v) {
#pragma unroll
  for (int off = 16; off >= 1; off >>= 1) v += __shfl_xor(v, off, 32);
  return v;
}

// ---------------------------------------------------------------------------
// WMMA fp32 GEMM:  C[M,N] = A[M,K] @ W[N,K]^T + bias[N]   (optional ReLU)
// Wave -> 16x32 C strip: two V_WMMA_F32_16X16X4_F32 per K-step sharing the
// A fragment (3 b64 loads / 2 WMMAs). Block = 4 waves = 32x64 C region.
// A tile (16x4): lane l holds M=l&15, K-pair k0+2*(l>>4)    (ISA 7.12.2)
// B tile (4x16): lane l holds N=l&15, K-pair k0+2*(l>>4)    (B[k][n] = W[n][k])
// C tile: vgpr r -> row m0+r+8*(l>>4), col n0+(l&15)
// ---------------------------------------------------------------------------
template <bool RELU>
__global__ __launch_bounds__(128) void gemm_bias_kernel(
    const float* __restrict__ A, const float* __restrict__ Wt,
    const float* __restrict__ bias, float* __restrict__ C,
    int M, int N, int K) {
  const int lane = threadIdx.x & 31;
  const int wid  = threadIdx.x >> 5;                 // 0..3
  const int m0   = blockIdx.y * 32 + (wid & 1) * 16;
  const int n0   = blockIdx.x * 64 + (wid >> 1) * 32;
  const int half = lane >> 4;                        // 0/1 -> K-pair select
  const int l15  = lane & 15;
  const float* arow  = A  + (size_t)(m0 + l15) * K;
  const float* brow0 = Wt + (size_t)(n0 + l15) * K;
  const float* brow1 = Wt + (size_t)(n0 + 16 + l15) * K;
  v8f acc0 = {}, acc1 = {};
  for (int k0 = 0; k0 < K; k0 += 4) {
    const int kk = k0 + (half << 1);
    v2f a  = *(const v2f*)(arow + kk);
    v2f b0 = *(const v2f*)(brow0 + kk);
    v2f b1 = *(const v2f*)(brow1 + kk);
    acc0 = __builtin_amdgcn_wmma_f32_16x16x4_f32(false, a, false, b0,
                                                 (short)0, acc0, false, false);
    acc1 = __builtin_amdgcn_wmma_f32_16x16x4_f32(false, a, false, b1,
                                                 (short)0, acc1, false, false);
  }
  const int col0 = n0 + l15, col1 = n0 + 16 + l15;
  const float bs0 = bias[col0], bs1 = bias[col1];
#pragma unroll
  for (int r = 0; r < 8; ++r) {
    int row = m0 + r + half * 8;
    float v0 = acc0[r] + bs0;
    float v1 = acc1[r] + bs1;
    if (RELU) { v0 = v0 > 0.f ? v0 : 0.f; v1 = v1 > 0.f ? v1 : 0.f; }
    C[(size_t)row * N + col0] = v0;
    C[(size_t)row * N + col1] = v1;
  }
}

// ---------------------------------------------------------------------------
// r[j] = [sin(pos*invf), cos(pos*invf)], pos = 1023-j
// ---------------------------------------------------------------------------
__global__ __launch_bounds__(256) void pos_emb_kernel(float* __restrict__ r) {
  int j = blockIdx.x, t = threadIdx.x;               // t = frequency idx 0..255
  float pos  = (float)(NN - 1 - j);
  float invf = expf(-(2.0f * (float)t / (float)DM) * 9.210340371976184f); // ln 1e4
  float a = pos * invf;
  r[j * DM + t]       = sinf(a);
  r[j * DM + 256 + t] = cosf(a);
}

// ---------------------------------------------------------------------------
// Memory self-attention (64 RAW mem rows act as K=V, position 64 uses the
// projected k/v of the current token), fused W0 + residual + LayerNorm n1.
// ---------------------------------------------------------------------------
__global__ __launch_bounds__(256) void sa_fused_kernel(
    const float* __restrict__ h,   const float* __restrict__ qsa,
    const float* __restrict__ ksa, const float* __restrict__ vsa,
    const float* __restrict__ mem,
    const float* __restrict__ W0,  const float* __restrict__ b0,
    const float* __restrict__ g,   const float* __restrict__ beta,
    float* __restrict__ hout) {
  __shared__ float q_s[DM], k_s[DM], v_s[DM], x_s[DM], attn_s[DM];
  __shared__ float e_s[NH];
  __shared__ float red_s[2 * 256];
  const int b = blockIdx.x, t = threadIdx.x, lane = t & 31, w = t >> 5;
  const float scale = 0.17677669529663689f;          // 1/sqrt(32)
  q_s[t] = qsa[b * DM + t];  q_s[t + 256] = qsa[b * DM + t + 256];
  k_s[t] = ksa[b * DM + t];  k_s[t + 256] = ksa[b * DM + t + 256];
  v_s[t] = vsa[b * DM + t];  v_s[t + 256] = vsa[b * DM + t + 256];
  __syncthreads();
  const int h0 = t >> 5;            // head owning dim t        (0..7)
  const int h1 = (t + 256) >> 5;    // head owning dim t+256    (8..15)
  float acc0 = 0.f, acc1 = 0.f, den0 = 0.f, den1 = 0.f;
  for (int j = 0; j < MEMLEN; ++j) {
    const float* xp = mem + ((size_t)j * BB + b) * DM;
    x_s[t] = xp[t]; x_s[t + 256] = xp[t + 256];
    __syncthreads();
#pragma unroll
    for (int hh = 0; hh < 2; ++hh) {                 // wave w -> heads w, w+8
      int hd = w + hh * 8;
      float p = q_s[hd * DH + lane] * x_s[hd * DH + lane];
      p = wave_sum(p);
      float s = 10.f * tanhf(p * scale);
      if (lane == 0) e_s[hd] = __expf(s);
    }
    __syncthreads();
    acc0 += e_s[h0] * x_s[t];       den0 += e_s[h0];
    acc1 += e_s[h1] * x_s[t + 256]; den1 += e_s[h1];
    __syncthreads();
  }
#pragma unroll
  for (int hh = 0; hh < 2; ++hh) {                   // position 64
    int hd = w + hh * 8;
    float p = q_s[hd * DH + lane] * k_s[hd * DH + lane];
    p = wave_sum(p);
    float s = 10.f * tanhf(p * scale);
    if (lane == 0) e_s[hd] = __expf(s);
  }
  __syncthreads();
  acc0 += e_s[h0] * v_s[t];       den0 += e_s[h0];
  acc1 += e_s[h1] * v_s[t + 256]; den1 += e_s[h1];
  __syncthreads();
  attn_s[t] = acc0 / den0; attn_s[t + 256] = acc1 / den1;
  __syncthreads();
  float v0 = b0[t] + h[b * DM + t];
  float v1 = b0[t + 256] + h[b * DM + t + 256];
  for (int c = 0; c < DM; ++c) {
    v0 += W0[(size_t)t * DM + c] * attn_s[c];
    v1 += W0[(size_t)(t + 256) * DM + c] * attn_s[c];
  }
  red_s[t] = v0 + v1; red_s[256 + t] = v0 * v0 + v1 * v1;
  __syncthreads();
  for (int off = 128; off > 0; off >>= 1) {
    if (t < off) { red_s[t] += red_s[t + off]; red_s[256 + t] += red_s[256 + t + off]; }
    __syncthreads();
  }
  float m = red_s[0] * (1.f / DM);
  float var = red_s[256] * (1.f / DM) - m * m;
  float rstd = rsqrtf(var + 1e-5f);
  hout[b * DM + t]       = (v0 - m) * rstd * g[t] + beta[t];
  hout[b * DM + t + 256] = (v1 - m) * rstd * g[t + 256] + beta[t + 256];
}

// ---------------------------------------------------------------------------
// qkC[b,h,:] = sum_d (q+u)[b,h,d] * Wk_a[h*32+d,:] ; cA[b,h] = (q+u).bk_a
// ---------------------------------------------------------------------------
__global__ __launch_bounds__(256) void qfold_kernel(
    const float* __restrict__ qa, const float* __restrict__ u,
    const float* __restrict__ Wk, const float* __restrict__ bk,
    float* __restrict__ qkC, float* __restrict__ cA) {
  __shared__ float qu_s[DM];
  const int b = blockIdx.x, t = threadIdx.x;
  qu_s[t]       = qa[b * DM + t] + u[t];
  qu_s[t + 256] = qa[b * DM + t + 256] + u[t + 256];
  __syncthreads();
  for (int hh = 0; hh < NH; ++hh) {
    float a0 = 0.f, a1 = 0.f;
    for (int d = 0; d < DH; ++d) {
      float qv = qu_s[hh * DH + d];
      const float* wr = Wk + (size_t)(hh * DH + d) * DM;
      a0 += qv * wr[t];
      a1 += qv * wr[t + 256];
    }
    qkC[((size_t)b * NH + hh) * DM + t]       = a0;
    qkC[((size_t)b * NH + hh) * DM + t + 256] = a1;
  }
  if (t < NH) {
    float c = 0.f;
    for (int d = 0; d < DH; ++d) c += qu_s[t * DH + d] * bk[t * DH + d];
    cA[b * NH + t] = c;
  }
}

// ---------------------------------------------------------------------------
// BD[j,b,h] = sum_d (q+v)[b,h,d] * ra[j, h*32+d]
// ---------------------------------------------------------------------------
__global__ __launch_bounds__(256) void bd_kernel(
    const float* __restrict__ qa, const float* __restrict__ vp,
    const float* __restrict__ ra, float* __restrict__ BD) {
  __shared__ float ra_s[DM], v_s[DM];
  const int j = blockIdx.x, t = threadIdx.x;
  ra_s[t] = ra[(size_t)j * DM + t]; ra_s[t + 256] = ra[(size_t)j * DM + t + 256];
  v_s[t] = vp[t]; v_s[t + 256] = vp[t + 256];
  __syncthreads();
  const float* qrow = qa + (size_t)t * DM;           // t indexes batch b
  for (int hh = 0; hh < NH; ++hh) {
    float s = 0.f;
#pragma unroll
    for (int d = 0; d < DH; ++d)
      s += (qrow[hh * DH + d] + v_s[hh * DH + d]) * ra_s[hh * DH + d];
    BD[((size_t)j * BB + t) * NH + hh] = s;
  }
}

// ---------------------------------------------------------------------------
// Flash cross-attention over N=1024 positions, fused Wv/W0 + residual + LN n2.
// One block per batch b, 16 waves = 16 heads. h_enc[:,b,:] streamed once via a
// 4-slot LDS ring filled by GLOBAL_LOAD_ASYNC_TO_LDS (3-deep pipeline) when
// available; register-staged ring otherwise.
// Pipeline invariant: loads complete in issue order; at top of iter j the wave
// has issued loads 0..j+2, so ASYNCcnt<=2 guarantees slot j&3 is resident.
// ---------------------------------------------------------------------------
__global__ __launch_bounds__(512) void cross_attn_kernel(
    const float* __restrict__ h,    const float* __restrict__ henc,
    const float* __restrict__ qkC,  const float* __restrict__ cA,
    const float* __restrict__ BD,   const unsigned char* __restrict__ mask,
    const float* __restrict__ Wv,   const float* __restrict__ bv,
    const float* __restrict__ W0,   const float* __restrict__ b0,
    const float* __restrict__ g,    const float* __restrict__ beta,
    float* __restrict__ hout, float* __restrict__ probs,  // probs may be null
    float* __restrict__ den_out) {
  __shared__ float x_s[4][DM];                       // 8 KB ring
  __shared__ float e_s[NH];
  __shared__ float ctx_s[NH * DM];                   // 32 KB
  __shared__ float attn_s[DM];
  __shared__ float red_s[2 * 512];
  const int b = blockIdx.x, t = threadIdx.x, lane = t & 31, hw = t >> 5;
  const float scale = 0.17677669529663689f;
  float qk_r[16];
  {
    const float* qp = qkC + ((size_t)b * NH + hw) * DM + lane * 16;
#pragma unroll
    for (int i = 0; i < 16; ++i) qk_r[i] = qp[i];
  }
  const float cbias = cA[b * NH + hw];
  float ctx_acc[NH];
#pragma unroll
  for (int i = 0; i < NH; ++i) ctx_acc[i] = 0.f;
  float den = 0.f;                                   // uniform across wave hw

#if USE_ASYNC_LDS
#pragma unroll
  for (int pj = 0; pj < 3; ++pj)                     // prime 3-deep
    async_copy_b32(&x_s[pj][t], henc + ((size_t)pj * BB + b) * DM + t);
#else
  x_s[0][t] = henc[(size_t)b * DM + t];              // prime slot 0
#endif

  for (int j = 0; j < NN; ++j) {
    const int cur = j & 3;
#if USE_ASYNC_LDS
    __builtin_amdgcn_s_wait_asynccnt(2);             // slot `cur` resident
#endif
    __syncthreads();                                 // barrier A
    // ---- phase 1: prefetch issue + scores ----
#if USE_ASYNC_LDS
    if (j + 3 < NN)
      async_copy_b32(&x_s[(j + 3) & 3][t],
                     henc + ((size_t)(j + 3) * BB + b) * DM + t);
#else
    float xn = 0.f;
    if (j + 1 < NN) xn = henc[((size_t)(j + 1) * BB + b) * DM + t];
#endif
    float p = 0.f;
#pragma unroll
    for (int i = 0; i < 16; ++i) p += qk_r[i] * x_s[cur][lane * 16 + i];
    p = wave_sum(p);
    float s = (p + cbias + BD[((size_t)j * BB + b) * NH + hw]) * scale;
    s = 10.f * tanhf(s);
    float e = mask[(size_t)j * BB + b] ? 0.f : __expf(s);
    den += e;
    if (lane == 0) {
      e_s[hw] = e;
      if (probs) probs[((size_t)j * BB + b) * NH + hw] = e;  // normalized later
    }
    __syncthreads();                                 // barrier B
    // ---- phase 2: context accumulation ----
    float xv = x_s[cur][t];
#pragma unroll
    for (int hy = 0; hy < NH; ++hy) ctx_acc[hy] += e_s[hy] * xv;
#if !USE_ASYNC_LDS
    if (j + 1 < NN) x_s[(j + 1) & 3][t] = xn;
#endif
  }
  __syncthreads();                                   // protect e_s recycle
  if (lane == 0) { e_s[hw] = den; den_out[b * NH + hw] = den; }
  __syncthreads();
#pragma unroll
  for (int hy = 0; hy < NH; ++hy) ctx_s[hy * DM + t] = ctx_acc[hy] / e_s[hy];
  __syncthreads();
  {                                                  // attn = bv + Wv.ctx
    float s = bv[t];
    const float* wr = Wv + (size_t)t * DM;
    const float* cp = ctx_s + (t >> 5) * DM;
    for (int c = 0; c < DM; ++c) s += wr[c] * cp[c];
    attn_s[t] = s;
  }
  __syncthreads();
  float val = b0[t] + h[b * DM + t];
  {
    const float* wr = W0 + (size_t)t * DM;
    for (int c = 0; c < DM; ++c) val += wr[c] * attn_s[c];
  }
  red_s[t] = val; red_s[512 + t] = val * val;
  __syncthreads();
  for (int off = 256; off > 0; off >>= 1) {
    if (t < off) { red_s[t] += red_s[t + off]; red_s[512 + t] += red_s[512 + t + off]; }
    __syncthreads();
  }
  float m = red_s[0] * (1.f / DM);
  float var = red_s[512] * (1.f / DM) - m * m;
  float rstd = rsqrtf(var + 1e-5f);
  hout[b * DM + t] = (val - m) * rstd * g[t] + beta[t];
}

__global__ __launch_bounds__(256) void norm_probs_kernel(
    float* __restrict__ probs, const float* __restrict__ den) {
  const int j = blockIdx.x, b = threadIdx.x;
#pragma unroll
  for (int hh = 0; hh < NH; ++hh)
    probs[((size_t)j * BB + b) * NH + hh] /= den[b * NH + hh];
}

__global__ __launch_bounds__(256) void add_ln_kernel(
    const float* __restrict__ hin, const float* __restrict__ f2,
    const float* __restrict__ g, const float* __restrict__ beta,
    float* __restrict__ hout) {
  __shared__ float red_s[2 * 256];
  const int b = blockIdx.x, t = threadIdx.x;
  float v0 = hin[b * DM + t] + f2[b * DM + t];
  float v1 = hin[b * DM + t + 256] + f2[b * DM + t + 256];
  red_s[t] = v0 + v1; red_s[256 + t] = v0 * v0 + v1 * v1;
  __syncthreads();
  for (int off = 128; off > 0; off >>= 1) {
    if (t < off) { red_s[t] += red_s[t + off]; red_s[256 + t] += red_s[256 + t + off]; }
    __syncthreads();
  }
  float m = red_s[0] * (1.f / DM);
  float var = red_s[256] * (1.f / DM) - m * m;
  float rstd = rsqrtf(var + 1e-5f);
  hout[b * DM + t]       = (v0 - m) * rstd * g[t] + beta[t];
  hout[b * DM + t + 256] = (v1 - m) * rstd * g[t + 256] + beta[t + 256];
}

// ---------------------------------------------------------------------------
// Per-layer param tensor order (sorted keys within each layer dict):
enum {
  I_W0_a_b = 0, I_W0_a_w, I_W0_sa_b, I_W0_sa_w,
  I_Wk_a_b, I_Wk_a_w, I_Wk_sa_b, I_Wk_sa_w,
  I_Wq_a_b, I_Wq_a_w, I_Wq_sa_b, I_Wq_sa_w,
  I_Wr_a_b, I_Wr_a_w, I_Wv_a_b, I_Wv_a_w,
  I_Wv_sa_b, I_Wv_sa_w, I_l1_b, I_l1_w, I_l2_b, I_l2_w,
  I_n1_b, I_n1_g, I_n2_b, I_n2_g, I_n3_b, I_n3_g, PER_LAYER
};

extern "C" void kernel_launch(void* const* d_in, const int* in_sizes, int n_in,
                              void* d_out, int out_size, void* d_ws, size_t ws_size,
                              hipStream_t stream) {
  (void)in_sizes; (void)n_in; (void)out_size; (void)ws_size;
  const float* h_t  = (const float*)d_in[0];
  const float* henc = (const float*)d_in[1];
  const unsigned char* mask = (const unsigned char*)d_in[2];
  const int PBASE = 3;
  const float* u  = (const float*)d_in[PBASE + NLAYER * PER_LAYER + 0];
  const float* vp = (const float*)d_in[PBASE + NLAYER * PER_LAYER + 1];

  // workspace carve-up (floats)
  float* W = (float*)d_ws;
  float* h    = W;                 // 256*512
  float* q    = h    + BB * DM;
  float* kbuf = q    + BB * DM;
  float* vbuf = kbuf + BB * DM;
  float* rpos = vbuf + BB * DM;    // 1024*512
  float* ra   = rpos + NN * DM;    // 1024*512
  float* qkC  = ra   + NN * DM;    // 256*16*512
  float* cA   = qkC  + (size_t)BB * NH * DM;   // 256*16
  float* BD   = cA   + BB * NH;    // 1024*256*16
  float* den  = BD   + (size_t)NN * BB * NH;   // 256*16
  float* f1   = den  + BB * NH;    // 256*2048
  float* f2   = f1   + BB * 2048;  // 256*512
  float* out  = (float*)d_out;

  (void)hipMemcpyAsync(h, h_t, (size_t)BB * DM * sizeof(float),
                       hipMemcpyDeviceToDevice, stream);
  pos_emb_kernel<<<NN, 256, 0, stream>>>(rpos);

  for (int L = 0; L < NLAYER; ++L) {
    const int base = PBASE + L * PER_LAYER;
    auto P = [&](int idx) -> const float* { return (const float*)d_in[base + idx]; };
    const float* mem = (const float*)d_in[PBASE + NLAYER * PER_LAYER + 2 + L];

    // ---- stage A: memory self-attention ----
    gemm_bias_kernel<false><<<dim3(DM / 64, BB / 32), 128, 0, stream>>>(
        h, P(I_Wq_sa_w), P(I_Wq_sa_b), q, BB, DM, DM);
    gemm_bias_kernel<false><<<dim3(DM / 64, BB / 32), 128, 0, stream>>>(
        h, P(I_Wk_sa_w), P(I_Wk_sa_b), kbuf, BB, DM, DM);
    gemm_bias_kernel<false><<<dim3(DM / 64, BB / 32), 128, 0, stream>>>(
        h, P(I_Wv_sa_w), P(I_Wv_sa_b), vbuf, BB, DM, DM);
    sa_fused_kernel<<<BB, 256, 0, stream>>>(
        h, q, kbuf, vbuf, mem, P(I_W0_sa_w), P(I_W0_sa_b),
        P(I_n1_g), P(I_n1_b), h);

    // ---- stage B: TXL relative cross-attention ----
    gemm_bias_kernel<false><<<dim3(DM / 64, BB / 32), 128, 0, stream>>>(
        h, P(I_Wq_a_w), P(I_Wq_a_b), q, BB, DM, DM);
    gemm_bias_kernel<false><<<dim3(DM / 64, NN / 32), 128, 0, stream>>>(
        rpos, P(I_Wr_a_w), P(I_Wr_a_b), ra, NN, DM, DM);
    qfold_kernel<<<BB, 256, 0, stream>>>(q, u, P(I_Wk_a_w), P(I_Wk_a_b), qkC, cA);
    bd_kernel<<<NN, 256, 0, stream>>>(q, vp, ra, BD);
    cross_attn_kernel<<<BB, 512, 0, stream>>>(
        h, henc, qkC, cA, BD, mask,
        P(I_Wv_a_w), P(I_Wv_a_b), P(I_W0_a_w), P(I_W0_a_b),
        P(I_n2_g), P(I_n2_b), h, (L == NLAYER - 1) ? out : nullptr, den);

    // ---- stage C: FFN ----
    gemm_bias_kernel<true><<<dim3(2048 / 64, BB / 32), 128, 0, stream>>>(
        h, P(I_l1_w), P(I_l1_b), f1, BB, 2048, DM);
    gemm_bias_kernel<false><<<dim3(DM / 64, BB / 32), 128, 0, stream>>>(
        f1, P(I_l2_w), P(I_l2_b), f2, BB, DM, 2048);
    add_ln_kernel<<<BB, 256, 0, stream>>>(h, f2, P(I_n3_g), P(I_n3_b), h);
  }

  norm_probs_kernel<<<NN, 256, 0, stream>>>(out, den);
}